// MHCHyperConnections_62964220559758
// MI455X (gfx1250) — compile-verified
//
#include <hip/hip_runtime.h>
#include <math.h>

// ---------------------------------------------------------------------------
// MHC hyper-connections, fused single-pass kernel for MI455X (gfx1250).
//
//  state   : (4, 2048, 4, 1024) f32   -> tile of 16 tokens = contiguous 256KiB
//  residual: (4, 2048, 1024)    f32
//  W       : (24, 4096)         f32   (stays resident in 192MB L2)
//  out     : pre (8192,1024) ++ new_state (8192,4,1024), flat f32
//
//  One workgroup (256 thr = 8 wave32) handles 16 tokens:
//   A) stage state tile into LDS via GLOBAL_LOAD_ASYNC_TO_LDS_B128 (ASYNCcnt)
//   B) rmsnorm scale per token (wave reductions)
//   C) raw[16x24] = norm(state) x W^T via V_WMMA_F32_16X16X4_F32,
//      K split over the 8 waves, combined with ds_add_f32
//   D) sigmoid gates + 4x4 Sinkhorn (10 iters), 1 thread/token
//   E) stream mixing outputs (bandwidth-bound; NT stores keep L2 clean for W)
// ---------------------------------------------------------------------------

typedef float v2f __attribute__((ext_vector_type(2)));
typedef float v4f __attribute__((ext_vector_type(4)));
typedef float v8f __attribute__((ext_vector_type(8)));
typedef int   v4i __attribute__((ext_vector_type(4)));

#define NSTREAMS 4
#define NEMBD    1024
#define TOKENS   8192                 // B*T = 4*2048
#define FLATD    4096                 // NSTREAMS*NEMBD
#define TILE_T   16
#define LSTRIDE  4100                 // padded LDS row stride (floats): bank-spread
#define NCOEFF   24
#define PRE_TOTAL (TOKENS * NEMBD)    // offset of new_state in d_out

#if defined(__has_builtin)
#if __has_builtin(__builtin_amdgcn_global_load_async_to_lds_b128)
#define USE_ASYNC_LDS 1
#endif
#endif
#ifndef USE_ASYNC_LDS
#define USE_ASYNC_LDS 0
#endif

typedef __attribute__((address_space(1))) v4i gv4i;   // global int4*
typedef __attribute__((address_space(3))) v4i lv4i;   // LDS int4*

__launch_bounds__(256)
__global__ void mhc_fused_kernel(const float* __restrict__ state,
                                 const float* __restrict__ residual,
                                 const float* __restrict__ W,
                                 const float* __restrict__ alpha,
                                 const float* __restrict__ bias,
                                 float* __restrict__ out)
{
    __shared__ float smem[TILE_T * LSTRIDE];      // 262400 B raw state tile
    __shared__ float raw_acc[2][16][16];          // cross-wave WMMA accum
    __shared__ float scale_s[TILE_T];
    __shared__ float hpre_s[TILE_T][4];
    __shared__ float hpost_s[TILE_T][4];
    __shared__ float hres_s[TILE_T][16];

    const int tid  = threadIdx.x;
    const int lane = tid & 31;
    const int wave = tid >> 5;
    const int tok0 = blockIdx.x * TILE_T;

    // ---- Phase A: stage contiguous 16-token state tile into LDS ----------
    {
        // prefetch residual tile into cache (gfx1250 global_prefetch_b8)
        const float* rbase = residual + (size_t)tok0 * NEMBD;
        __builtin_prefetch(rbase + tid * 64, 0, 0);
        __builtin_prefetch(rbase + tid * 64 + 32, 0, 0);

        const v4f* g = (const v4f*)(state + (size_t)tok0 * FLATD);
#if USE_ASYNC_LDS
        // Direct global -> LDS async copy (no VGPR round trip), ASYNCcnt-tracked.
        #pragma unroll 4
        for (int it = 0; it < 64; ++it) {
            int idx = tid + it * 256;            // float4 index, 0..16383
            int m   = idx >> 10;                 // token within tile
            int k   = (idx & 1023) << 2;         // element within 4096
            gv4i* gp = (gv4i*)(uintptr_t)(g + idx);
            lv4i* lp = (lv4i*)(uint32_t)(uintptr_t)&smem[m * LSTRIDE + k];
            __builtin_amdgcn_global_load_async_to_lds_b128(gp, lp, 0, 0);
        }
#if __has_builtin(__builtin_amdgcn_s_wait_asynccnt)
        __builtin_amdgcn_s_wait_asynccnt(0);
#else
        asm volatile("s_wait_asynccnt 0" ::: "memory");
#endif
#else
        #pragma unroll 4
        for (int it = 0; it < 64; ++it) {
            int idx = tid + it * 256;            // float4 index, 0..16383
            int m   = idx >> 10;                 // token within tile
            int k   = (idx & 1023) << 2;         // element within 4096
            v4f v = g[idx];
            *(v4f*)&smem[m * LSTRIDE + k] = v;
        }
#endif
        // zero WMMA cross-wave accumulators (512 floats)
        ((float*)raw_acc)[tid]       = 0.0f;
        ((float*)raw_acc)[tid + 256] = 0.0f;
    }
    __syncthreads();

    // ---- Phase B: rmsnorm scale per token (2 tokens per wave) ------------
    for (int s = 0; s < 2; ++s) {
        int m = wave * 2 + s;
        const float* row = &smem[m * LSTRIDE];
        float ssq = 0.0f;
        #pragma unroll
        for (int it = 0; it < 32; ++it) {
            v4f v = *(const v4f*)&row[(lane + it * 32) << 2];
            ssq += v.x * v.x + v.y * v.y + v.z * v.z + v.w * v.w;
        }
        ssq += __shfl_xor(ssq, 16, 32);
        ssq += __shfl_xor(ssq,  8, 32);
        ssq += __shfl_xor(ssq,  4, 32);
        ssq += __shfl_xor(ssq,  2, 32);
        ssq += __shfl_xor(ssq,  1, 32);
        if (lane == 0)
            scale_s[m] = rsqrtf(ssq * (1.0f / FLATD) + 1.1920929e-07f);
    }
    __syncthreads();

    // ---- Phase C: raw[16x24] = norm(flat) x W^T via f32 WMMA -------------
    // A (16x4 per step): lane L<16 holds A[M=L][k0],A[M=L][k1];
    //                    lane L>=16 holds A[M=L-16][k2],A[M=L-16][k3]
    // B (4x16 per step): N = lane&15, same K split by lane-half.
    // D: VGPR r, lane L -> raw[M = r + 8*(L>=16)][N = L&15]
    {
        const int   row  = lane & 15;
        const int   hi   = lane >> 4;
        const int   col1 = 16 + row;                       // second N-tile col
        const float msk1 = (col1 < NCOEFF) ? 1.0f : 0.0f;  // mask pad cols
        const float* w0  = W + (size_t)row * FLATD;
        const float* w1  = W + (size_t)((col1 < NCOEFF) ? col1 : (NCOEFF - 1)) * FLATD;
        const float  sc  = scale_s[row];
        const float* xr  = &smem[row * LSTRIDE];

        v8f acc0 = {};
        v8f acc1 = {};
        const int ksBase = wave * 512;                     // K split over waves
        #pragma unroll 4
        for (int k4 = 0; k4 < 512; k4 += 4) {
            const int kk = ksBase + k4 + 2 * hi;
            v2f a, b0, b1;
            a[0]  = xr[kk]     * sc;
            a[1]  = xr[kk + 1] * sc;
            b0[0] = w0[kk];
            b0[1] = w0[kk + 1];
            b1[0] = w1[kk]     * msk1;
            b1[1] = w1[kk + 1] * msk1;
            acc0 = __builtin_amdgcn_wmma_f32_16x16x4_f32(false, a, false, b0,
                                                         (short)0, acc0, false, false);
            acc1 = __builtin_amdgcn_wmma_f32_16x16x4_f32(false, a, false, b1,
                                                         (short)0, acc1, false, false);
        }
        #pragma unroll
        for (int r = 0; r < 8; ++r) {
            const int M = r + hi * 8;
            atomicAdd(&raw_acc[0][M][row], acc0[r]);       // ds_add_f32
            atomicAdd(&raw_acc[1][M][row], acc1[r]);
        }
    }
    __syncthreads();

    // ---- Phase D: gates + Sinkhorn (1 thread per token) ------------------
    if (tid < TILE_T) {
        const float a0 = alpha[0], a1 = alpha[1], a2 = alpha[2];
        float raw[NCOEFF];
        #pragma unroll
        for (int c = 0; c < 16; ++c)       raw[c] = raw_acc[0][tid][c];
        #pragma unroll
        for (int c = 16; c < NCOEFF; ++c)  raw[c] = raw_acc[1][tid][c - 16];

        #pragma unroll
        for (int n = 0; n < 4; ++n) {
            float zp = a0 * raw[n] + bias[n];
            hpre_s[tid][n]  = 1.0f / (1.0f + expf(-zp));
            float zq = a1 * raw[4 + n] + bias[4 + n];
            hpost_s[tid][n] = 2.0f / (1.0f + expf(-zq));
        }

        float mm[16];
        #pragma unroll
        for (int e = 0; e < 16; ++e) {
            float z = a2 * raw[8 + e] + bias[8 + e];
            z = fminf(fmaxf(z, -20.0f), 20.0f);
            mm[e] = expf(z);
        }
        #pragma unroll
        for (int it = 0; it < 10; ++it) {
            #pragma unroll
            for (int j = 0; j < 4; ++j) {                  // column normalize
                float cs  = mm[j] + mm[4 + j] + mm[8 + j] + mm[12 + j];
                float inv = 1.0f / fmaxf(cs, 1e-12f);
                mm[j] *= inv; mm[4 + j] *= inv; mm[8 + j] *= inv; mm[12 + j] *= inv;
            }
            #pragma unroll
            for (int i = 0; i < 4; ++i) {                  // row normalize
                float rs  = mm[4 * i] + mm[4 * i + 1] + mm[4 * i + 2] + mm[4 * i + 3];
                float inv = 1.0f / fmaxf(rs, 1e-12f);
                mm[4 * i] *= inv; mm[4 * i + 1] *= inv; mm[4 * i + 2] *= inv; mm[4 * i + 3] *= inv;
            }
        }
        #pragma unroll
        for (int e = 0; e < 16; ++e) hres_s[tid][e] = mm[e];
    }
    __syncthreads();

    // ---- Phase E: mixing + outputs (bandwidth-bound streaming) -----------
    for (int s = 0; s < 2; ++s) {
        const int m   = wave * 2 + s;
        const int tok = tok0 + m;
        const float* srow = &smem[m * LSTRIDE];

        float hp[4], q[4], R[4][4];
        #pragma unroll
        for (int n = 0; n < 4; ++n) { hp[n] = hpre_s[m][n]; q[n] = hpost_s[m][n]; }
        #pragma unroll
        for (int i = 0; i < 4; ++i)
            #pragma unroll
            for (int j = 0; j < 4; ++j) R[i][j] = hres_s[m][4 * i + j];

        const v4f* resg = (const v4f*)(residual + (size_t)tok * NEMBD);
        v4f* preo = (v4f*)(out + (size_t)tok * NEMBD);
        v4f* nso  = (v4f*)(out + (size_t)PRE_TOTAL + (size_t)tok * FLATD);

        #pragma unroll
        for (int it = 0; it < 8; ++it) {
            const int d4 = lane + it * 32;                 // float4 index < 256
            v4f s0 = *(const v4f*)&srow[0 * NEMBD + d4 * 4];
            v4f s1 = *(const v4f*)&srow[1 * NEMBD + d4 * 4];
            v4f s2 = *(const v4f*)&srow[2 * NEMBD + d4 * 4];
            v4f s3 = *(const v4f*)&srow[3 * NEMBD + d4 * 4];
            v4f rv = __builtin_nontemporal_load(resg + d4);

            v4f p = hp[0] * s0 + hp[1] * s1 + hp[2] * s2 + hp[3] * s3;
            __builtin_nontemporal_store(p, preo + d4);     // written once, NT

            #pragma unroll
            for (int i = 0; i < 4; ++i) {
                v4f acc = q[i] * rv;
                acc += R[i][0] * s0;
                acc += R[i][1] * s1;
                acc += R[i][2] * s2;
                acc += R[i][3] * s3;
                __builtin_nontemporal_store(acc, nso + i * (NEMBD / 4) + d4);
            }
        }
    }
}

extern "C" void kernel_launch(void* const* d_in, const int* in_sizes, int n_in,
                              void* d_out, int out_size, void* d_ws, size_t ws_size,
                              hipStream_t stream) {
    (void)in_sizes; (void)n_in; (void)d_ws; (void)ws_size; (void)out_size;
    const float* state    = (const float*)d_in[0];
    const float* residual = (const float*)d_in[1];
    const float* W        = (const float*)d_in[2];
    const float* alpha    = (const float*)d_in[3];
    const float* bias     = (const float*)d_in[4];
    float* out            = (float*)d_out;

    dim3 grid(TOKENS / TILE_T);   // 512 workgroups, one 16-token tile each
    dim3 block(256);              // 8 wave32
    hipLaunchKernelGGL(mhc_fused_kernel, grid, block, 0, stream,
                       state, residual, W, alpha, bias, out);
}